// MultiLabel_LossFunc_40346922779396
// MI455X (gfx1250) — compile-verified
//
#include <hip/hip_runtime.h>

// BP-MLL pairwise ranking loss for [B=64, L=188, V=8192].
// HBM-bound: 789 MB streamed once -> ~34 us floor at 23.3 TB/s.
// One workgroup (256 thr = 8 wave32) per (b,l) row of 8192 elements.
// One v_exp_f32 per element (pos/neg split is branch-free cndmask routing).
// Wave cross-lane reduction runs through V_WMMA_F32_16X16X4_F32.

typedef __attribute__((ext_vector_type(2))) float v2f;
typedef __attribute__((ext_vector_type(8))) float v8f;

#define BB   64
#define LLEN 188
#define VV   8192
#define TPB  256
#define NROWS (BB * LLEN)          // 12032
#define ITERS (VV / (TPB * 4))     // 8 float4 loads per thread

__global__ __launch_bounds__(TPB)
void bpmll_row_kernel(const float* __restrict__ scores,
                      const int*   __restrict__ labels,
                      const int*   __restrict__ lengths,
                      float*       __restrict__ row_out)
{
    const int row = blockIdx.x;          // 0 .. NROWS-1
    const int b   = row / LLEN;
    const int l   = row - b * LLEN;
    const long base = (long)row * VV;

    const float4* __restrict__ sp = reinterpret_cast<const float4*>(scores + base);
    const int4*   __restrict__ yp = reinterpret_cast<const int4*>(labels + base);

    float pos_acc = 0.0f;
    float neg_acc = 0.0f;

    // Fully coalesced: wave w, iter i touches a contiguous 512B span.
    #pragma unroll
    for (int i = 0; i < ITERS; ++i) {
        const int idx = i * TPB + threadIdx.x;   // float4 index
        const float4 s = sp[idx];
        const int4   y = yp[idx];
        // one transcendental per element; route to pos or neg via cndmask
        { float e = __expf(y.x ? s.x : -s.x); pos_acc += y.x ? e : 0.0f; neg_acc += y.x ? 0.0f : e; }
        { float e = __expf(y.y ? s.y : -s.y); pos_acc += y.y ? e : 0.0f; neg_acc += y.y ? 0.0f : e; }
        { float e = __expf(y.z ? s.z : -s.z); pos_acc += y.z ? e : 0.0f; neg_acc += y.z ? 0.0f : e; }
        { float e = __expf(y.w ? s.w : -s.w); pos_acc += y.w ? e : 0.0f; neg_acc += y.w ? 0.0f : e; }
    }

    // ---- wave32 cross-lane reduction via the matrix pipe ----
    // A (16x4 f32, 2 VGPRs): VGPR0 = acc (K0 lanes0-15 / K2 lanes16-31), VGPR1 = 0.
    // B = all-ones (layout invariant) => C[m][n] = acc[m] + acc[m+16] for every n.
    v2f a_pos; a_pos[0] = pos_acc; a_pos[1] = 0.0f;
    v2f a_neg; a_neg[0] = neg_acc; a_neg[1] = 0.0f;
    v2f b_one; b_one[0] = 1.0f;    b_one[1] = 1.0f;
    v8f c0 = {};
    v8f cpos = __builtin_amdgcn_wmma_f32_16x16x4_f32(
        false, a_pos, false, b_one, (short)0, c0, false, false);
    v8f cneg = __builtin_amdgcn_wmma_f32_16x16x4_f32(
        false, a_neg, false, b_one, (short)0, c0, false, false);

    // lanes 0-15 hold C[m][lane] for m=0..7 in v0..v7; lanes 16-31 hold m=8..15.
    float p8 = cpos[0]+cpos[1]+cpos[2]+cpos[3]+cpos[4]+cpos[5]+cpos[6]+cpos[7];
    float n8 = cneg[0]+cneg[1]+cneg[2]+cneg[3]+cneg[4]+cneg[5]+cneg[6]+cneg[7];
    const float pos_wave = p8 + __shfl_xor(p8, 16, 32);  // full 32-lane sum, all lanes
    const float neg_wave = n8 + __shfl_xor(n8, 16, 32);

    __shared__ float spos[TPB / 32];
    __shared__ float sneg[TPB / 32];
    const int wave = threadIdx.x >> 5;
    if ((threadIdx.x & 31) == 0) { spos[wave] = pos_wave; sneg[wave] = neg_wave; }
    __syncthreads();

    if (threadIdx.x == 0) {
        float P = 0.0f, N = 0.0f;
        #pragma unroll
        for (int w = 0; w < TPB / 32; ++w) { P += spos[w]; N += sneg[w]; }
        const float m = (l < lengths[b]) ? 1.0f : 0.0f;
        row_out[row] = m * P * N;
    }
}

// Deterministic fixed-order final reduction of the 12032 per-row products.
__global__ __launch_bounds__(TPB)
void bpmll_reduce_kernel(const float* __restrict__ row_vals,
                         float*       __restrict__ out)
{
    __shared__ float sm[TPB];
    float acc = 0.0f;
    for (int i = threadIdx.x; i < NROWS; i += TPB) acc += row_vals[i];
    sm[threadIdx.x] = acc;
    __syncthreads();
    #pragma unroll
    for (int s = TPB / 2; s > 0; s >>= 1) {
        if (threadIdx.x < s) sm[threadIdx.x] += sm[threadIdx.x + s];
        __syncthreads();
    }
    if (threadIdx.x == 0) {
        // total * V / B  =  total * 8192 / 64  =  total * 128
        out[0] = sm[0] * ((float)VV / (float)BB);
    }
}

extern "C" void kernel_launch(void* const* d_in, const int* in_sizes, int n_in,
                              void* d_out, int out_size, void* d_ws, size_t ws_size,
                              hipStream_t stream) {
    const float* scores  = (const float*)d_in[0];   // [B, L, V] f32
    const int*   labels  = (const int*)d_in[1];     // [B, L, V] i32
    const int*   lengths = (const int*)d_in[2];     // [B] i32
    float*       out     = (float*)d_out;           // [1] f32
    float*       ws      = (float*)d_ws;            // >= NROWS floats scratch

    bpmll_row_kernel<<<NROWS, TPB, 0, stream>>>(scores, labels, lengths, ws);
    bpmll_reduce_kernel<<<1, TPB, 0, stream>>>(ws, out);
}